// ObjectLoss_72421738545221
// MI455X (gfx1250) — compile-verified
//
#include <hip/hip_runtime.h>

typedef float v2f __attribute__((ext_vector_type(2)));
typedef float v8f __attribute__((ext_vector_type(8)));

#define EPS_ 1e-7f
#define B_ 16
#define A_ 3
#define T_ 64
#define G0 64
#define G1 32
#define G2 16
#define N0 (B_*A_*G0*G0)   /* 196608 */
#define N1 (B_*A_*G1*G1)   /*  49152 */
#define N2 (B_*A_*G2*G2)   /*  12288 */
#define NTOT (N0+N1+N2)    /* 258048 */
#define W0 (N0/32)         /* 6144 words */
#define W1 (N1/32)         /* 1536 words */
#define W2 (N2/32)         /*  384 words */
#define NWORDS (W0+W1+W2)  /* 8064 words = 32256 B */

#define ELEMS_PER_THREAD 4
#define NTHREADS (NTOT/ELEMS_PER_THREAD)   /* 64512 = 252*256 */

// DS_SWIZZLE_B32 group-of-32 XOR exchange: offset = {xor[4:0]<<10 | or<<5 | and}
#define SWIZ_XOR(x, m) \
  __int_as_float(__builtin_amdgcn_ds_swizzle(__float_as_int(x), (((m) << 10) | 0x1f)))

// ---------------------------------------------------------------------------
// Pass 0: clear gt bitmap (workspace) and the output scalar.
// ---------------------------------------------------------------------------
__global__ void ObjLoss_zero_kernel(unsigned* __restrict__ bmp, float* __restrict__ out) {
  int i = blockIdx.x * blockDim.x + threadIdx.x;
  if (i < NWORDS) bmp[i] = 0u;
  if (i == 0) out[0] = 0.0f;
}

// ---------------------------------------------------------------------------
// Pass 1: target -> gt bitmap scatter (1024 targets, 3 scales each).
// atomicOr dedups targets mapping to the same cell (matches .at[].max(mask)).
// ---------------------------------------------------------------------------
__device__ __forceinline__ void scatter_scale(float tx, float ty, float fw, float fh,
                                              int b, int g,
                                              const float* __restrict__ anc,
                                              unsigned* __restrict__ bmp) {
  const float tw = fw * (float)g;
  const float th = fh * (float)g;
  float best = -1.0f;
  int ba = 0;
#pragma unroll
  for (int a = 0; a < 3; ++a) {
    const float aw = anc[2 * a + 0];
    const float ah = anc[2 * a + 1];
    const float inter = fminf(aw, tw) * fminf(ah, th);
    const float uni = aw * ah + tw * th - inter;
    const float iou = __fdividef(inter, uni);     // rcp-based; ample for >0.5 test
    if (iou > best) { best = iou; ba = a; }       // first-max tie rule (jnp.argmax)
  }
  if (best > 0.5f) {
    int ii = (int)floorf(tx * (float)g);
    int jj = (int)floorf(ty * (float)g);
    ii = min(g - 1, max(0, ii));
    jj = min(g - 1, max(0, jj));
    const unsigned idx = ((unsigned)((b * 3 + ba) * g + jj)) * (unsigned)g + (unsigned)ii;
    atomicOr(&bmp[idx >> 5], 1u << (idx & 31u));
  }
}

__global__ void ObjLoss_scatter_kernel(const float* __restrict__ tgt,
                                       const float* __restrict__ anc0,
                                       const float* __restrict__ anc1,
                                       const float* __restrict__ anc2,
                                       unsigned* __restrict__ bmp) {
  const int t = blockIdx.x * blockDim.x + threadIdx.x;  // 0..1023 == B_*T_
  const int b = t >> 6;                                 // T_ == 64
  const float* p = tgt + (size_t)t * 5;
  const float tx = p[1], ty = p[2], fw = p[3], fh = p[4];
  scatter_scale(tx, ty, fw, fh, b, G0, anc0, bmp);
  scatter_scale(tx, ty, fw, fh, b, G1, anc1, bmp + W0);
  scatter_scale(tx, ty, fw, fh, b, G2, anc2, bmp + W0 + W1);
}

// ---------------------------------------------------------------------------
// Pass 2: fused BCE + reduction. 4 consecutive elements per thread:
//  - all 4 stay inside one scale (N0, N1 are multiples of 4)
//  - all 4 gt bits live in the same bitmap word (one load)
//  - 4 independent strided global loads in flight per thread (latency hiding)
// BCE branch-free: r = bit ? q : (1-q); term = -log(r). ONE native log per
// element, no divergent dual transcendental paths.
// Cross-lane sum on the matrix unit: V_WMMA_F32_16X16X4_F32 with all-ones A
// and B = {acc0, acc1} sums 64 partials (2/lane) into replicated column sums;
// 4 single-instruction DS_SWIZZLE XOR steps finish 16 columns -> scalar.
// ---------------------------------------------------------------------------
__global__ void ObjLoss_loss_kernel(const float* __restrict__ out0,
                                    const float* __restrict__ out1,
                                    const float* __restrict__ out2,
                                    const unsigned* __restrict__ bmp,
                                    float* __restrict__ loss) {
  const int tid = blockIdx.x * blockDim.x + threadIdx.x;   // < NTHREADS exactly
  const int e0 = tid * ELEMS_PER_THREAD;                   // first element, 4-aligned

  const float* base;
  const unsigned* bm;
  unsigned l;
  float invN;
  if (e0 < N0) {
    base = out0; bm = bmp;           l = (unsigned)e0;             invN = 1.0f / (float)N0;
  } else if (e0 < N0 + N1) {
    base = out1; bm = bmp + W0;      l = (unsigned)(e0 - N0);      invN = 1.0f / (float)N1;
  } else {
    base = out2; bm = bmp + W0 + W1; l = (unsigned)(e0 - N0 - N1); invN = 1.0f / (float)N2;
  }

  // one bitmap word covers bits l .. l+3 (l is 4-aligned)
  const unsigned word = bm[l >> 5];
  const unsigned sh = l & 31u;

  float p[ELEMS_PER_THREAD];
#pragma unroll
  for (int k = 0; k < ELEMS_PER_THREAD; ++k)        // 4 loads issued back-to-back
    p[k] = base[(size_t)(l + k) * 85 + 4];          // channel 4 of 85

  float s0 = 0.0f, s1 = 0.0f;                       // sums of +log(r); negate later
#pragma unroll
  for (int k = 0; k < ELEMS_PER_THREAD; ++k) {
    const float q = fminf(fmaxf(p[k], EPS_), 1.0f - EPS_);
    const bool bit = (word >> (sh + k)) & 1u;
    const float r = bit ? q : (1.0f - q);           // select BEFORE the log
    const float lg = __logf(r);                     // native v_log_f32 path
    if (k & 1) s1 += lg; else s0 += lg;
  }
  const float acc0 = s0 * -invN;                    // per-scale mean scaling
  const float acc1 = s1 * -invN;

  // --- wave32 reduction on the matrix unit (EXEC all-ones: exact tiling) ---
  v2f a;  a.x = 1.0f; a.y = 1.0f;     // A = ones(16x4)
  v2f bb; bb.x = acc0; bb.y = acc1;   // B rows 0..3 = 64 per-lane partials
  v8f c = {};
  c = __builtin_amdgcn_wmma_f32_16x16x4_f32(false, a, false, bb, (short)0, c,
                                            false, false);
  float v = c[0];                     // lane L: colsum(L%16), replicated in halves
  v += SWIZ_XOR(v, 1);
  v += SWIZ_XOR(v, 2);
  v += SWIZ_XOR(v, 4);
  v += SWIZ_XOR(v, 8);                // every lane: full 32-lane (64-partial) sum

  __shared__ float wsum[8];
  const int wave = threadIdx.x >> 5;
  if ((threadIdx.x & 31) == 0) wsum[wave] = v;
  __syncthreads();
  if (threadIdx.x == 0) {
    float s = 0.0f;
#pragma unroll
    for (int w = 0; w < 8; ++w) s += wsum[w];
    atomicAdd(loss, s);
  }
}

// ---------------------------------------------------------------------------
extern "C" void kernel_launch(void* const* d_in, const int* in_sizes, int n_in,
                              void* d_out, int out_size, void* d_ws, size_t ws_size,
                              hipStream_t stream) {
  // setup_inputs() dict order: out0, anchors0, out1, anchors1, out2, anchors2, targets
  const float* out0 = (const float*)d_in[0];
  const float* anc0 = (const float*)d_in[1];
  const float* out1 = (const float*)d_in[2];
  const float* anc1 = (const float*)d_in[3];
  const float* out2 = (const float*)d_in[4];
  const float* anc2 = (const float*)d_in[5];
  const float* tgt  = (const float*)d_in[6];
  float* loss = (float*)d_out;
  unsigned* bmp = (unsigned*)d_ws;   // 8064 words = 32256 bytes of workspace

  ObjLoss_zero_kernel<<<(NWORDS + 255) / 256, 256, 0, stream>>>(bmp, loss);
  ObjLoss_scatter_kernel<<<(B_ * T_) / 256, 256, 0, stream>>>(tgt, anc0, anc1, anc2, bmp);
  ObjLoss_loss_kernel<<<NTHREADS / 256, 256, 0, stream>>>(out0, out1, out2, bmp, loss);
}